// EmbeddingLayer_3229815407304
// MI455X (gfx1250) — compile-verified
//
#include <hip/hip_runtime.h>
#include <hip/hip_bf16.h>

#ifndef __has_builtin
#define __has_builtin(x) 0
#endif

#if __has_builtin(__builtin_amdgcn_global_load_async_to_lds_b128) && \
    __has_builtin(__builtin_amdgcn_s_wait_asynccnt)
#define EMB_USE_ASYNC_LDS 1
#else
#define EMB_USE_ASYNC_LDS 0
#endif

namespace {

constexpr int   kD       = 1024;   // d_model
constexpr int   kS       = 2048;   // sequence length
constexpr int   kB       = 8;      // batch
constexpr int   kThreads = 256;    // kD / 4 floats per thread
constexpr float kScale   = 1.0f / 0.9f;           // inverted-dropout scale, p = 0.1
// PE: inv_freq(i) = 10000^(-2i/D) = 2^(i * (-2*log2(10000)/D))
constexpr float kLog2_10000 = 13.2877123795494f;
constexpr float kCoef       = -2.0f * kLog2_10000 / (float)kD;
constexpr float kInv2Pi     = 0.15915494309189535f;
constexpr float kTwoPi      = 6.283185307179586f;

// Native 16B vector type (HIP's float4 is a struct; builtins need real vectors).
typedef float v4f __attribute__((ext_vector_type(4)));

#if EMB_USE_ASYNC_LDS
// Builtin signature (from hipcc diagnostic): params are
//   (int4 __device__* /*AS1*/, int4 __shared__* /*AS3*/, Imm int offset, Imm int cpol)
typedef int v4i __attribute__((vector_size(16)));
typedef __attribute__((address_space(1))) v4i* g_v4i_ptr;
typedef __attribute__((address_space(3))) v4i* l_v4i_ptr;
#endif

// sin/cos with cheap explicit range reduction: |ang| < 2048 rad here, so a
// single divide-free mod-2pi keeps __sincosf's hardware v_sin_f32/v_cos_f32
// in-domain and avoids libm's Payne-Hanek giant-argument path (hundreds of
// VALU ops + v_mul_u64 chains).
__device__ __forceinline__ void fast_sincos(float ang, float* s, float* c) {
  float r = ang * kInv2Pi;
  r -= floorf(r);               // [0, 1)
  __sincosf(r * kTwoPi, s, c);  // in [0, 2pi): hardware fast path is valid
}

// Deterministic per-element hash (stateless, replay-safe). ~PCG finisher mix.
__device__ __forceinline__ unsigned hash32(unsigned x) {
  x ^= x >> 17;  x *= 0xED5AD4BBu;
  x ^= x >> 11;  x *= 0xAC4C1B51u;
  x ^= x >> 15;  x *= 0x31848BABu;
  x ^= x >> 14;
  return x;
}

__device__ __forceinline__ float drop(float v, unsigned h, int byte_lane) {
  // keep probability 230/256 ~= 0.8984 ~ (1 - 0.1)
  const unsigned b = (h >> (8 * byte_lane)) & 0xFFu;
  return (b < 230u) ? v * kScale : 0.0f;
}

} // namespace

// out[b,s,d] = dropout( emb[x[b,s], d] + PE[s,d] )
// grid = S blocks, block = 256 threads; each thread owns a float4 column slice
// and iterates the 8 batch rows (8 independent 16B gathers in flight).
__global__ __launch_bounds__(kThreads) void emb_pe_dropout_kernel(
    const int* __restrict__ tok,        // [B, S] token ids
    const float* __restrict__ emb,      // [NUM_TOKENS, D]
    float* __restrict__ out)            // [B, S, D]
{
  const int s   = (int)blockIdx.x;
  const int tid = (int)threadIdx.x;

  // ---- token ids for this column: block-uniform -> scalar loads ----
  int idx[kB];
#pragma unroll
  for (int b = 0; b < kB; ++b) idx[b] = tok[b * kS + s];

  v4f row[kB];

#if EMB_USE_ASYNC_LDS
  // CDNA5 async data mover path: 8 in-flight GLOBAL_LOAD_ASYNC_TO_LDS_B128
  // per wave, then a single s_wait_asynccnt 0. Each lane reads back only its
  // own 16B slot, so ASYNCcnt==0 (async LDS write complete) is sufficient
  // ordering; no workgroup barrier required.
  __shared__ v4f stage[kB * kThreads];   // 32 KB of 320 KB/WGP
#pragma unroll
  for (int b = 0; b < kB; ++b) {
    const v4f* gp = (const v4f*)(emb + (size_t)idx[b] * kD) + tid;
    v4f*       lp = &stage[b * kThreads + tid];
    __builtin_amdgcn_global_load_async_to_lds_b128(
        (g_v4i_ptr)(void*)gp,
        (l_v4i_ptr)(void*)lp,
        /*offset=*/0, /*cpol=*/0);
  }
#endif

  // ---- sinusoidal PE for (s, 4*tid .. 4*tid+3): overlap with async copies ----
  float pe0, pe1, pe2, pe3;
  {
    const float fi0 = (float)(2 * tid);
    const float fi1 = (float)(2 * tid + 1);
    const float a0  = (float)s * exp2f(fi0 * kCoef);
    const float a1  = (float)s * exp2f(fi1 * kCoef);
    fast_sincos(a0, &pe0, &pe1);   // d = 4*tid   (sin), 4*tid+1 (cos)
    fast_sincos(a1, &pe2, &pe3);   // d = 4*tid+2 (sin), 4*tid+3 (cos)
  }

#if EMB_USE_ASYNC_LDS
  __builtin_amdgcn_s_wait_asynccnt(0);
#pragma unroll
  for (int b = 0; b < kB; ++b) row[b] = stage[b * kThreads + tid];
#else
  // Fallback: direct 16B vector gathers (8 independent loads in flight).
#pragma unroll
  for (int b = 0; b < kB; ++b)
    row[b] = ((const v4f*)(emb + (size_t)idx[b] * kD))[tid];
#endif

  // ---- add PE, dropout, non-temporal 16B stores (output never re-read;
  // keep L2 free for the reused embedding rows) ----
#pragma unroll
  for (int b = 0; b < kB; ++b) {
    const unsigned vid = (unsigned)((b * kS + s) * kThreads + tid);
    const unsigned h   = hash32(vid);
    v4f v = row[b];
    v.x = drop(v.x + pe0, h, 0);
    v.y = drop(v.y + pe1, h, 1);
    v.z = drop(v.z + pe2, h, 2);
    v.w = drop(v.w + pe3, h, 3);
    __builtin_nontemporal_store(v, (v4f*)out + (size_t)(b * kS + s) * kThreads + tid);
  }
}

extern "C" void kernel_launch(void* const* d_in, const int* in_sizes, int n_in,
                              void* d_out, int out_size, void* d_ws, size_t ws_size,
                              hipStream_t stream) {
  (void)in_sizes; (void)n_in; (void)out_size; (void)d_ws; (void)ws_size;
  const int*   tok = (const int*)d_in[0];     // x: [8, 2048] int32
  const float* emb = (const float*)d_in[1];   // token_embeddings: [50257, 1024] f32
  float*       out = (float*)d_out;           // [8, 2048, 1024] f32

  emb_pe_dropout_kernel<<<kS, kThreads, 0, stream>>>(tok, emb, out);
}